// GAT_Layer_54185307406515
// MI455X (gfx1250) — compile-verified
//
#include <hip/hip_runtime.h>

// ---------------------------------------------------------------------------
// GATConv (B=4, N=10000, E=160000, D=128, H=4, O=64) for gfx1250 (MI455X).
// GEMM uses exact-fp32 V_WMMA_F32_16X16X4_F32 with K-pair-interleaved LDS so
// each B operand is a single ds_load_b64 into an even-aligned VGPR pair.
// Edge scatter exploits the 192MB L2: the 41MB [N,B,H,O] accumulator is fully
// L2-resident so the 164M f32 atomic adds resolve in L2, not HBM.
// ---------------------------------------------------------------------------

#define NB 10000
#define BB 4
#define EE 160000
#define DD 128
#define HH 4
#define OO 64
#define HO 256          // H*O
#define MROWS (BB * NB) // 40000, multiple of 16
#define NEG_SLOPE 0.2f

typedef float v2f __attribute__((ext_vector_type(2)));
typedef float v8f __attribute__((ext_vector_type(8)));

// order-preserving float<->uint mapping so segment-max can use native u32 atomics
__device__ __forceinline__ unsigned fenc(float f) {
  unsigned u = __float_as_uint(f);
  return (u & 0x80000000u) ? ~u : (u | 0x80000000u);
}
__device__ __forceinline__ float fdec(unsigned u) {
  return __uint_as_float((u & 0x80000000u) ? (u ^ 0x80000000u) : ~u);
}

// --------------------------- WMMA projection GEMM ---------------------------
// h[40000 x 256] = x[40000 x 128] * W[128 x 256], fp32 WMMA 16x16x4.
// Block: 256 thr (8 waves) -> 128 rows. W staged in LDS in two 64-row K-halves,
// stored as K-pair-interleaved float2 so B operands are single b64 LDS loads.
__global__ __launch_bounds__(256) void gemm_h(const float* __restrict__ x,
                                              const float* __restrict__ W,
                                              float* __restrict__ h) {
  __shared__ float2 Wlds[32 * HO]; // 32 K-pairs x 256 cols x 8B = 64 KB
  const int tid = threadIdx.x;
  const int lane = tid & 31;
  const int wave = tid >> 5;
  const int rowBase = blockIdx.x * 128 + wave * 16;
  const bool active = rowBase < MROWS;

  v8f acc[16];
  const v8f vzero = {0.f, 0.f, 0.f, 0.f, 0.f, 0.f, 0.f, 0.f};
#pragma unroll
  for (int t = 0; t < 16; ++t) acc[t] = vzero;

  const int m = lane & 15;        // A-matrix row owned by this lane
  const int kb = (lane >> 4) << 1; // K sub-pair: lanes 0-15 -> K{0,1}, 16-31 -> K{2,3}
  const int kp = lane >> 4;        // pair offset within a 4-wide K step

  for (int kblk = 0; kblk < DD; kblk += 64) {
    // stage W[kblk..kblk+63][0..255] as interleaved K-pairs; thread owns col=tid
#pragma unroll 4
    for (int p = 0; p < 32; ++p) {
      float2 pr;
      pr.x = W[(size_t)(kblk + 2 * p) * HO + tid];
      pr.y = W[(size_t)(kblk + 2 * p + 1) * HO + tid];
      Wlds[p * HO + tid] = pr;
    }
    __syncthreads();

    if (active) {
      const float* xrow = x + (size_t)(rowBase + m) * DD + kblk;
      for (int kk = 0; kk < 64; kk += 4) {
        v2f a;
        a.x = xrow[kk + kb];
        a.y = xrow[kk + kb + 1];
        const float2* w0 = Wlds + ((kk >> 1) + kp) * HO + (lane & 15);
#pragma unroll
        for (int t = 0; t < 16; ++t) {
          float2 bp = w0[t * 16]; // {W[kb][c], W[kb+1][c]} in one b64 load
          v2f b;
          b.x = bp.x;
          b.y = bp.y;
          acc[t] = __builtin_amdgcn_wmma_f32_16x16x4_f32(
              false, a, false, b, (short)0, acc[t], false, false);
        }
      }
    }
    __syncthreads();
  }

  if (active) {
    const int nLocal = lane & 15;
    const int rOff = (lane >> 4) * 8; // C layout: VGPR v -> rows v / v+8
#pragma unroll
    for (int t = 0; t < 16; ++t)
#pragma unroll
      for (int v = 0; v < 8; ++v)
        h[(size_t)(rowBase + rOff + v) * HO + t * 16 + nLocal] = acc[t][v];
  }
}

// ------------------- per-(b,n,head) attention coefficients -------------------
__global__ __launch_bounds__(256) void calc_attn_node(
    const float* __restrict__ h, const float* __restrict__ att_src,
    const float* __restrict__ att_dst, float* __restrict__ a_src,
    float* __restrict__ a_dst) {
  const int wv = blockIdx.x * 8 + (threadIdx.x >> 5); // (b*N+n)*H + head
  if (wv >= MROWS * HH) return;
  const int lane = threadIdx.x & 31;
  const int head = wv & 3;
  const float2 hv = ((const float2*)(h + (size_t)wv * OO))[lane];
  const float2 as = ((const float2*)(att_src + head * OO))[lane];
  const float2 ad = ((const float2*)(att_dst + head * OO))[lane];
  float vs = hv.x * as.x + hv.y * as.y;
  float vd = hv.x * ad.x + hv.y * ad.y;
#pragma unroll
  for (int off = 16; off > 0; off >>= 1) { // wave32 reduction
    vs += __shfl_xor(vs, off);
    vd += __shfl_xor(vd, off);
  }
  if (lane == 0) { a_src[wv] = vs; a_dst[wv] = vd; }
}

// coef[h] = sum_o W_edge[h,o]*att_edge[h,o]
__global__ void calc_edge_coef(const float* __restrict__ W_edge,
                               const float* __restrict__ att_edge,
                               float* __restrict__ coef) {
  const int t = threadIdx.x;
  if (t < HH) {
    float s = 0.f;
    for (int o = 0; o < OO; ++o) s += W_edge[t * OO + o] * att_edge[t * OO + o];
    coef[t] = s;
  }
}

__device__ __forceinline__ float edge_alpha(const float* a_src, const float* a_dst,
                                            int sN, int dN, int b, int hh,
                                            float ae, const float* coef) {
  float a = a_src[((size_t)b * NB + sN) * HH + hh] +
            a_dst[((size_t)b * NB + dN) * HH + hh] + ae * coef[hh];
  return a > 0.f ? a : a * NEG_SLOPE;
}

// ------------------------ segment max (u32 atomic max) -----------------------
__global__ __launch_bounds__(256) void edge_max(
    const int* __restrict__ src, const int* __restrict__ dst,
    const float* __restrict__ eattr, const float* __restrict__ a_src,
    const float* __restrict__ a_dst, const float* __restrict__ coef,
    unsigned* __restrict__ mbuf) {
  const int e = blockIdx.x * 256 + threadIdx.x;
  if (e >= EE) return;
  const int sN = src[e], dN = dst[e];
  const float ae = eattr[e];
  for (int b = 0; b < BB; ++b)
#pragma unroll
    for (int hh = 0; hh < HH; ++hh) {
      float a = edge_alpha(a_src, a_dst, sN, dN, b, hh, ae, coef);
      atomicMax(&mbuf[(size_t)dN * 16 + b * 4 + hh], fenc(a));
    }
}

// --------------------------- segment sum of exp ------------------------------
__global__ __launch_bounds__(256) void edge_sum(
    const int* __restrict__ src, const int* __restrict__ dst,
    const float* __restrict__ eattr, const float* __restrict__ a_src,
    const float* __restrict__ a_dst, const float* __restrict__ coef,
    const unsigned* __restrict__ mbuf, float* __restrict__ sden) {
  const int e = blockIdx.x * 256 + threadIdx.x;
  if (e >= EE) return;
  const int sN = src[e], dN = dst[e];
  const float ae = eattr[e];
  for (int b = 0; b < BB; ++b)
#pragma unroll
    for (int hh = 0; hh < HH; ++hh) {
      float a = edge_alpha(a_src, a_dst, sN, dN, b, hh, ae, coef);
      const size_t mi = (size_t)dN * 16 + b * 4 + hh;
      atomicAdd(&sden[mi], __expf(a - fdec(mbuf[mi])));
    }
}

// ------------------- weighted message scatter (block/edge) -------------------
// 1024 floats per edge; gather from h[b,src] and atomic-add into acc[dst,b]
// are both contiguous 256-float chunks -> fully coalesced; acc lives in L2.
__global__ __launch_bounds__(256) void edge_scatter(
    const int* __restrict__ src, const int* __restrict__ dst,
    const float* __restrict__ eattr, const float* __restrict__ a_src,
    const float* __restrict__ a_dst, const float* __restrict__ coef,
    const unsigned* __restrict__ mbuf, const float* __restrict__ sden,
    const float* __restrict__ h, float* __restrict__ acc) {
  __shared__ float wsh[16];
  const int e = blockIdx.x;
  const int t = threadIdx.x;
  const int sN = src[e], dN = dst[e];
  if (t < 16) {
    const int b = t >> 2, hh = t & 3;
    float a = edge_alpha(a_src, a_dst, sN, dN, b, hh, eattr[e], coef);
    const size_t mi = (size_t)dN * 16 + b * 4 + hh;
    float p = __expf(a - fdec(mbuf[mi]));
    wsh[t] = p / fmaxf(sden[mi], 1e-16f);
  }
  __syncthreads();
#pragma unroll
  for (int i = t; i < BB * HO; i += 256) {
    const int b = i >> 8, rem = i & 255;
    float val = wsh[b * 4 + (rem >> 6)] * h[((size_t)b * NB + sN) * HO + rem];
    atomicAdd(&acc[((size_t)dN * 4 + b) * HO + rem], val);
  }
}

// ------------------ mean over heads + bias + [N,B]->[B,N] --------------------
__global__ __launch_bounds__(256) void finalize(const float* __restrict__ acc,
                                                const float* __restrict__ bias,
                                                float* __restrict__ out) {
  const int i = blockIdx.x * 256 + threadIdx.x;
  if (i >= BB * NB * OO) return;
  const int o = i & 63;
  const int bn = i >> 6;
  const int n = bn % NB, b = bn / NB;
  const float* p = acc + ((size_t)n * 4 + b) * HO + o;
  out[i] = (p[0] + p[64] + p[128] + p[192]) * 0.25f + bias[o];
}

// ---------------------------------------------------------------------------
extern "C" void kernel_launch(void* const* d_in, const int* in_sizes, int n_in,
                              void* d_out, int out_size, void* d_ws, size_t ws_size,
                              hipStream_t stream) {
  const float* x        = (const float*)d_in[0];
  const int*   eidx     = (const int*)d_in[1];
  const float* eattr    = (const float*)d_in[2];
  const float* W        = (const float*)d_in[3];
  const float* att_src  = (const float*)d_in[4];
  const float* att_dst  = (const float*)d_in[5];
  const float* W_edge   = (const float*)d_in[6];
  const float* att_edge = (const float*)d_in[7];
  const float* bias     = (const float*)d_in[8];
  float* out = (float*)d_out;

  char* ws = (char*)d_ws;
  // workspace layout (bytes)
  float*    h_buf  = (float*)   (ws + 0);          // 40,960,000  [B,N,H,O]
  float*    a_src  = (float*)   (ws + 40960000);   //  2,560,000  [B,N,H]
  float*    a_dst  = (float*)   (ws + 43520000);   //  2,560,000  [B,N,H]
  unsigned* mbuf   = (unsigned*)(ws + 46080000);   //  2,560,000  [N,B,H] enc-max
  float*    sden   = (float*)   (ws + 48640000);   //  2,560,000  [N,B,H]
  float*    coef   = (float*)   (ws + 51200000);   //         16  [H]
  float*    acc    = (float*)   (ws + 51200256);   // 40,960,000  [N,B,H,O]

  const int* srcI = eidx;
  const int* dstI = eidx + EE;

  // zero m, sden, coef pad, acc in one contiguous memset (graph-capturable)
  hipMemsetAsync(ws + 46080000, 0, 46080256, stream);

  calc_edge_coef<<<1, 64, 0, stream>>>(W_edge, att_edge, coef);
  gemm_h<<<(MROWS + 127) / 128, 256, 0, stream>>>(x, W, h_buf);
  calc_attn_node<<<(MROWS * HH) / 8, 256, 0, stream>>>(h_buf, att_src, att_dst,
                                                       a_src, a_dst);
  edge_max<<<(EE + 255) / 256, 256, 0, stream>>>(srcI, dstI, eattr, a_src,
                                                 a_dst, coef, mbuf);
  edge_sum<<<(EE + 255) / 256, 256, 0, stream>>>(srcI, dstI, eattr, a_src,
                                                 a_dst, coef, mbuf, sden);
  edge_scatter<<<EE, 256, 0, stream>>>(srcI, dstI, eattr, a_src, a_dst, coef,
                                       mbuf, sden, h_buf, acc);
  finalize<<<(BB * NB * OO + 255) / 256, 256, 0, stream>>>(acc, bias, out);
}